// DeformationGraph_7567732376324
// MI455X (gfx1250) — compile-verified
//
#include <hip/hip_runtime.h>
#include <math.h>

// CDNA5 (gfx1250) wave32 WMMA fragment types for V_WMMA_F32_16X16X4_F32:
//   A: 16x4 f32  -> 2 VGPRs (v2f),  B: 4x16 f32 -> 2 VGPRs,  C/D: 16x16 -> 8 VGPRs
typedef __attribute__((ext_vector_type(2))) float v2f;
typedef __attribute__((ext_vector_type(8))) float v8f;
typedef __attribute__((ext_vector_type(4))) unsigned int u32x4;
typedef __attribute__((ext_vector_type(8))) int i32x8;
typedef __attribute__((ext_vector_type(4))) int i32x4;

// Address-space pointer types for async/TDM ops (probe: src param is AS(1) int*)
typedef __attribute__((address_space(1))) int g_int;
typedef __attribute__((address_space(3))) int l_int;

// ---- CDNA5 feature tiers (evaluated per compilation pass; host pass -> plain) ----
#if defined(__gfx1250__)
#if __has_builtin(__builtin_amdgcn_global_load_async_to_lds_b32)
#define HAVE_ASYNC_LDS 1
#endif
#if __has_builtin(__builtin_amdgcn_tensor_load_to_lds)
#define HAVE_TDM 1
#if defined(__clang_major__) && (__clang_major__ >= 23)
#define TDM_6ARG 1   // clang-23 lane: 6-arg (u32x4,i32x8,i32x4,i32x4,i32x8,i32)
#endif
#endif
#endif

__device__ __forceinline__ void wait_tensorcnt0() {
#if __has_builtin(__builtin_amdgcn_s_wait_tensorcnt)
  __builtin_amdgcn_s_wait_tensorcnt(0);
#else
  asm volatile("s_wait_tensorcnt 0x0" ::: "memory");
#endif
}

__device__ __forceinline__ void wait_asynccnt0() {
#if __has_builtin(__builtin_amdgcn_s_wait_asynccnt)
  __builtin_amdgcn_s_wait_asynccnt(0);
#else
  asm volatile("s_wait_asynccnt 0x0" ::: "memory");
#endif
}

// ---------------------------------------------------------------------------
// Rodrigues exactly as the reference.
// ---------------------------------------------------------------------------
__device__ __forceinline__ void rodrigues9(float ax, float ay, float az, float* R) {
  float ex = ax + 1e-8f, ey = ay + 1e-8f, ez = az + 1e-8f;
  float ang = sqrtf(ex * ex + ey * ey + ez * ez);
  float inv = 1.0f / ang;
  float ux = ax * inv, uy = ay * inv, uz = az * inv;
  float h = 0.5f * ang;
  float w = cosf(h), s = sinf(h);
  float qx = s * ux, qy = s * uy, qz = s * uz;
  float qn = rsqrtf(w * w + qx * qx + qy * qy + qz * qz);
  w *= qn; qx *= qn; qy *= qn; qz *= qn;
  float w2 = w * w, x2 = qx * qx, y2 = qy * qy, z2 = qz * qz;
  float wx = w * qx, wy = w * qy, wz = w * qz;
  float xy = qx * qy, xz = qx * qz, yz = qy * qz;
  R[0] = w2 + x2 - y2 - z2; R[1] = 2.f * xy - 2.f * wz; R[2] = 2.f * wy + 2.f * xz;
  R[3] = 2.f * wz + 2.f * xy; R[4] = w2 - x2 + y2 - z2; R[5] = 2.f * yz - 2.f * wx;
  R[6] = 2.f * xz - 2.f * wy; R[7] = 2.f * wx + 2.f * yz; R[8] = w2 - x2 - y2 + z2;
}

// ---------------------------------------------------------------------------
// Kernel 1: h = relu(feat @ W1 + b1), feat[b*J+j] = [nodes[j], cond[b]]
// One wave per 16x16 output tile; K-loop of f32 16x16x4 WMMAs. Wave-uniform
// control flow so EXEC stays all-ones around WMMA.
// ---------------------------------------------------------------------------
__device__ __forceinline__ float featval(const float* __restrict__ nodes,
                                         const float* __restrict__ cond,
                                         int C, int b, int j, int k) {
  return (k < 3) ? nodes[3 * j + k] : cond[b * C + (k - 3)];
}

__global__ __launch_bounds__(256) void mlp1_wmma_kernel(
    const float* __restrict__ nodes, const float* __restrict__ cond,
    const float* __restrict__ W1, const float* __restrict__ b1,
    float* __restrict__ hbuf, int B, int Jn, int C, int H) {
  const int wv   = blockIdx.x * (blockDim.x >> 5) + (threadIdx.x >> 5);
  const int lane = threadIdx.x & 31;
  const int ntiles = H >> 4;
  const int mtiles = (B * Jn) >> 4;
  if (wv >= mtiles * ntiles) return;           // wave-uniform guard
  const int mt = wv / ntiles, nt = wv % ntiles;
  const int lcol = lane & 15, half = lane >> 4;
  const int row = mt * 16 + lcol;
  const int b = row / Jn, j = row % Jn;
  const int col = nt * 16 + lcol;
  const int kfull = 3 + C;                     // 72, divisible by 4

  v8f acc = {};
  for (int kk = 0; kk < kfull; kk += 4) {
    const int ka = kk + 2 * half;
    v2f a, bb;
    a[0] = featval(nodes, cond, C, b, j, ka);
    a[1] = featval(nodes, cond, C, b, j, ka + 1);
    bb[0] = W1[(size_t)ka * H + col];
    bb[1] = W1[(size_t)(ka + 1) * H + col];
    acc = __builtin_amdgcn_wmma_f32_16x16x4_f32(false, a, false, bb,
                                                (short)0, acc, false, false);
  }
  const float bias = b1[col];
#pragma unroll
  for (int v = 0; v < 8; ++v) {
    const int m = mt * 16 + v + 8 * half;
    hbuf[(size_t)m * H + col] = fmaxf(acc[v] + bias, 0.0f);
  }
}

// ---------------------------------------------------------------------------
// Kernel 2: per-joint tf6 = h @ W2 + b2; Rodrigues; compose with root;
// scale + translate. Writes T as (B*J, 16) row-major 4x4.
// ---------------------------------------------------------------------------
__global__ __launch_bounds__(256) void finish_transforms_kernel(
    const float* __restrict__ hbuf, const float* __restrict__ W2,
    const float* __restrict__ b2, const float* __restrict__ root,
    const float* __restrict__ trans, const float* __restrict__ scale,
    float* __restrict__ T, int B, int Jn, int H) {
  const int row = blockIdx.x * blockDim.x + threadIdx.x;
  if (row >= B * Jn) return;
  const int b = row / Jn;

  float o[6];
#pragma unroll
  for (int c = 0; c < 6; ++c) o[c] = b2[c];
  const float* hr = hbuf + (size_t)row * H;
  for (int n = 0; n < H; ++n) {
    const float hv = hr[n];
#pragma unroll
    for (int c = 0; c < 6; ++c) o[c] += hv * W2[n * 6 + c];
  }

  float R[9], Rr[9];
  rodrigues9(o[0], o[1], o[2], R);
  rodrigues9(root[3 * b + 0], root[3 * b + 1], root[3 * b + 2], Rr);

  float M[12];
#pragma unroll
  for (int i = 0; i < 3; ++i) {
#pragma unroll
    for (int k = 0; k < 3; ++k)
      M[i * 4 + k] = Rr[i * 3 + 0] * R[0 * 3 + k] + Rr[i * 3 + 1] * R[1 * 3 + k] +
                     Rr[i * 3 + 2] * R[2 * 3 + k];
    M[i * 4 + 3] = Rr[i * 3 + 0] * o[3] + Rr[i * 3 + 1] * o[4] + Rr[i * 3 + 2] * o[5];
  }
  const float s = scale[b];
  float* Td = T + (size_t)row * 16;
#pragma unroll
  for (int i = 0; i < 3; ++i)
#pragma unroll
    for (int k = 0; k < 4; ++k) Td[i * 4 + k] = M[i * 4 + k] * s;
  Td[3]  += trans[3 * b + 0] * s;
  Td[7]  += trans[3 * b + 1] * s;
  Td[11] += trans[3 * b + 2] * s;
  Td[12] = 0.f; Td[13] = 0.f; Td[14] = 0.f; Td[15] = 1.f;
}

// ---------------------------------------------------------------------------
// Kernel 3 (dominant): fused distance-GEMM + streaming top-5 + skinning.
// One wave per 16 points. Score s = n.x - 0.5|n|^2 (max s == nearest).
// Staging: block's 128 points async-copied into LDS (ASYNCcnt); raw nodes
// DMA'd by the Tensor Data Mover from wave 0 (TENSORcnt); -0.5|n|^2 computed
// into a second LDS array; block's T-slice warmed via global_prefetch.
// ---------------------------------------------------------------------------
#define JMAX 1024
#define KNN 5

__device__ __forceinline__ void insert5(float s, int idx, float* ts, int* ti) {
  const bool ins = s > ts[4];
  ts[4] = ins ? s : ts[4];
  ti[4] = ins ? idx : ti[4];
#pragma unroll
  for (int k = 4; k > 0; --k) {
    const bool sw = ts[k] > ts[k - 1];
    const float fs = ts[k - 1]; const int fi = ti[k - 1];
    ts[k - 1] = sw ? ts[k] : fs; ti[k - 1] = sw ? ti[k] : fi;
    ts[k]     = sw ? fs : ts[k]; ti[k]     = sw ? fi : ti[k];
  }
}

__global__ __launch_bounds__(256) void knn_skin_kernel(
    const float* __restrict__ x, const float* __restrict__ nodes,
    const float* __restrict__ T, float* __restrict__ out,
    int B, int N, int Jn) {
  __shared__ float s_nodes3[3 * JMAX];  // raw xyz (TDM-staged)
  __shared__ float s_nn[JMAX];          // -0.5*|n|^2
  __shared__ float s_pts[8 * 16 * 3];   // this block's points (async-staged)

  const int waves_pb = blockDim.x >> 5;
  const int base3 = blockIdx.x * waves_pb * 48;     // first x float of block
  const int npts3 = min(waves_pb * 48, 3 * B * N - base3);

  // (a) async-stage this block's points into LDS
#if defined(HAVE_ASYNC_LDS)
  for (int i = threadIdx.x; i < npts3; i += blockDim.x)
    __builtin_amdgcn_global_load_async_to_lds_b32(
        (g_int*)(x + base3 + i), (l_int*)&s_pts[i], 0, 0);
#else
  for (int i = threadIdx.x; i < npts3; i += blockDim.x) s_pts[i] = x[base3 + i];
#endif

  // (b) warm this block's T-slice (64KB) in cache for the gather phase
  {
    const int bblk = (base3 / 3) / N;  // batch of first point (uniform per block)
    const float* Tb = T + (size_t)bblk * Jn * 16;
    __builtin_prefetch(Tb + (size_t)threadIdx.x * 32, 0, 3);
    __builtin_prefetch(Tb + (size_t)(threadIdx.x + blockDim.x) * 32, 0, 3);
  }

  // (c) stage raw nodes into LDS via TDM (wave 0), async, or plain loads
#if defined(HAVE_TDM)
  if (threadIdx.x < 32) {  // wave 0 issues the DMA (EXEC ignored by TDM)
    const unsigned long long ga = (unsigned long long)(const void*)nodes;
    const unsigned lds_base =
        (unsigned)(unsigned long long)(__attribute__((address_space(3))) float*)&s_nodes3[0];
    const unsigned nelem = 3u * (unsigned)Jn;  // 3072 f32 elements
    u32x4 g0;
    g0[0] = 1u;                                   // count=1 (valid user D#)
    g0[1] = lds_base;                             // lds_addr [63:32]
    g0[2] = (unsigned)(ga & 0xffffffffu);         // global_addr[31:0]
    g0[3] = (unsigned)((ga >> 32) & 0x1ffffffu)   // global_addr[56:32]
            | 0x80000000u;                        // type=2 ("image")
    i32x8 g1;
    g1[0] = 0x00020000;                           // data_size=2 (4 bytes)
    g1[1] = (int)((nelem & 0xffffu) << 16);       // tensor_dim0[15:0]
    g1[2] = (int)(((nelem >> 16) & 0xffffu)       // tensor_dim0[31:16]
                  | (1u << 16));                  // tensor_dim1 = 1
    g1[3] = (int)((nelem & 0xffffu) << 16);       // tile_dim0 = nelem
    g1[4] = 0;                                    // tile_dim1/2 unused
    g1[5] = (int)nelem;                           // tensor_dim0_stride[31:0]
    g1[6] = 0;
    g1[7] = 0;
    i32x4 gz = {0, 0, 0, 0};                      // groups 2/3: <=2D tensor
#if defined(TDM_6ARG)
    i32x8 gz8 = {0, 0, 0, 0, 0, 0, 0, 0};
    __builtin_amdgcn_tensor_load_to_lds(g0, g1, gz, gz, gz8, 0);
#else
    __builtin_amdgcn_tensor_load_to_lds(g0, g1, gz, gz, 0);
#endif
    wait_tensorcnt0();
  }
#elif defined(HAVE_ASYNC_LDS)
  for (int j = threadIdx.x; j < 3 * Jn; j += blockDim.x)
    __builtin_amdgcn_global_load_async_to_lds_b32(
        (g_int*)(nodes + j), (l_int*)&s_nodes3[j], 0, 0);
#else
  for (int j = threadIdx.x; j < 3 * Jn; j += blockDim.x) s_nodes3[j] = nodes[j];
#endif

#if defined(HAVE_ASYNC_LDS)
  wait_asynccnt0();  // each wave waits for its own async stores into LDS
#endif
  __syncthreads();

  for (int j = threadIdx.x; j < Jn; j += blockDim.x) {
    const float n0 = s_nodes3[3 * j + 0], n1 = s_nodes3[3 * j + 1], n2 = s_nodes3[3 * j + 2];
    s_nn[j] = -0.5f * (n0 * n0 + n1 * n1 + n2 * n2);
  }
  __syncthreads();

  const int wlocal = threadIdx.x >> 5;
  const int tile = blockIdx.x * waves_pb + wlocal;
  const int tiles = (B * N) >> 4;
  if (tile >= tiles) return;                 // wave-uniform: EXEC all-ones below
  const int lane = threadIdx.x & 31;
  const int lcol = lane & 15, half = lane >> 4;
  const int prow = tile * 16 + lcol;         // this lane's point (global b*N+p)
  const int b = prow / N;

  // B-fragment: lane holds xh[point lcol][k], k = j + 2*half, j in {0,1}
  const float px = s_pts[wlocal * 48 + 3 * lcol + 0];
  const float py = s_pts[wlocal * 48 + 3 * lcol + 1];
  const float pz = s_pts[wlocal * 48 + 3 * lcol + 2];
  const float xh[4] = {px, py, pz, 1.0f};
  v2f bfrag;
  bfrag[0] = xh[2 * half + 0];
  bfrag[1] = xh[2 * half + 1];
  const float xx = px * px + py * py + pz * pz;

  float ts[KNN]; int ti[KNN];
#pragma unroll
  for (int k = 0; k < KNN; ++k) { ts[k] = -INFINITY; ti[k] = 0; }

  for (int t = 0; t < Jn >> 4; ++t) {
    // A-fragment: lane holds node (16t + lcol): half 0 -> comps {0,1},
    // half 1 -> {n2, -0.5|n|^2}
    const int nid = t * 16 + lcol;
    v2f afrag;
    afrag[0] = half ? s_nodes3[3 * nid + 2] : s_nodes3[3 * nid + 0];
    afrag[1] = half ? s_nn[nid]             : s_nodes3[3 * nid + 1];
    v8f acc = {};
    acc = __builtin_amdgcn_wmma_f32_16x16x4_f32(false, afrag, false, bfrag,
                                                (short)0, acc, false, false);
    // Prefilter: skip the sorted-insert when no lane can improve its top-5
    // (compiler emits s_cbranch_execz to skip the whole block).
    const float m = fmaxf(fmaxf(fmaxf(acc[0], acc[1]), fmaxf(acc[2], acc[3])),
                          fmaxf(fmaxf(acc[4], acc[5]), fmaxf(acc[6], acc[7])));
    if (m > ts[4]) {
      const int base = t * 16 + half * 8;    // D rows owned by this lane
#pragma unroll
      for (int v = 0; v < 8; ++v) insert5(acc[v], base + v, ts, ti);
    }
  }

  // merge the two half-lane candidate lists for each point
#pragma unroll
  for (int k = 0; k < KNN; ++k) {
    const float os = __shfl_xor(ts[k], 16, 32);
    const int   oi = __shfl_xor(ti[k], 16, 32);
    insert5(os, oi, ts, ti);
  }

  if (half == 0) {
    float w[KNN], wsum = 0.f;
#pragma unroll
    for (int k = 0; k < KNN; ++k) {
      const float d2 = fmaxf(xx - 2.0f * ts[k], 0.0f);
      const float dist = sqrtf(d2);
      w[k] = -logf(fminf(dist, 1.0f) - 1e-6f);
      wsum += w[k];
    }
    const float winv = 1.0f / wsum;
    float o0 = 0.f, o1 = 0.f, o2 = 0.f;
#pragma unroll
    for (int k = 0; k < KNN; ++k) {
      const float* Tm = T + ((size_t)b * Jn + ti[k]) * 16;
      const float wk = w[k] * winv;
      o0 += wk * (Tm[0] * px + Tm[1] * py + Tm[2]  * pz + Tm[3]);
      o1 += wk * (Tm[4] * px + Tm[5] * py + Tm[6]  * pz + Tm[7]);
      o2 += wk * (Tm[8] * px + Tm[9] * py + Tm[10] * pz + Tm[11]);
    }
    out[3 * prow + 0] = o0;
    out[3 * prow + 1] = o1;
    out[3 * prow + 2] = o2;
  }
}

// ---------------------------------------------------------------------------
// Host launcher
// ---------------------------------------------------------------------------
extern "C" void kernel_launch(void* const* d_in, const int* in_sizes, int n_in,
                              void* d_out, int out_size, void* d_ws, size_t ws_size,
                              hipStream_t stream) {
  const float* x     = (const float*)d_in[0];
  const float* cond  = (const float*)d_in[1];
  const float* nodes = (const float*)d_in[2];
  const float* root  = (const float*)d_in[3];
  const float* trans = (const float*)d_in[4];
  const float* scale = (const float*)d_in[5];
  const float* W1    = (const float*)d_in[6];
  const float* b1    = (const float*)d_in[7];
  const float* W2    = (const float*)d_in[8];
  const float* b2    = (const float*)d_in[9];
  float* out = (float*)d_out;

  const int B  = in_sizes[3] / 3;          // smpl_root_orient (B,3)
  const int Jn = in_sizes[2] / 3;          // nodes (J,3)
  const int C  = in_sizes[1] / B;          // cond_smpl (B,C)
  const int N  = in_sizes[0] / (3 * B);    // x (B,N,3)
  const int H  = in_sizes[7];              // b1 (H,)

  float* hbuf = (float*)d_ws;                       // B*J*H f32 (4 MB)
  float* T    = hbuf + (size_t)B * Jn * H;          // B*J*16 f32 (256 KB)

  {
    const int waves = ((B * Jn) >> 4) * (H >> 4);
    const int blocks = (waves + 7) / 8;
    mlp1_wmma_kernel<<<blocks, 256, 0, stream>>>(nodes, cond, W1, b1, hbuf, B, Jn, C, H);
  }
  {
    const int total = B * Jn;
    finish_transforms_kernel<<<(total + 255) / 256, 256, 0, stream>>>(
        hbuf, W2, b2, root, trans, scale, T, B, Jn, H);
  }
  {
    const int tiles = (B * N) >> 4;
    const int blocks = (tiles + 7) / 8;
    knn_skin_kernel<<<blocks, 256, 0, stream>>>(x, nodes, T, out, B, N, Jn);
  }
}